// BiLSTMEmbed_64063732187514
// MI455X (gfx1250) — compile-verified
//
#include <hip/hip_runtime.h>
#include <cmath>

// Problem dims (match reference)
#define SEQ    2048
#define EMB    512
#define UNITS  512
#define GATES  2048          // 4*UNITS
// Scan decomposition
#define DIRBLK 32            // workgroups per direction
#define UPB    16            // units per scan block
#define CPB    64            // gate-columns per scan block (4 gates * UPB)
#define WHS_STRIDE 513       // LDS padding: (c*513 + r) % 64 == (c + r) % 64 -> conflict free

typedef float v2f __attribute__((ext_vector_type(2)));
typedef float v8f __attribute__((ext_vector_type(8)));

// ---------------------------------------------------------------------------
// Kernel A: embedding gather  embeds[l,:] = vocab[idx[l],:]
// ---------------------------------------------------------------------------
__global__ void gather_embeds(const int* __restrict__ idx,
                              const float* __restrict__ vocab,
                              float* __restrict__ embeds) {
  const int l = blockIdx.x;
  const int e = threadIdx.x * 4;                       // 128 thr * 4 = 512
  const float4 v = *reinterpret_cast<const float4*>(vocab + (size_t)idx[l] * EMB + e);
  *reinterpret_cast<float4*>(embeds + (size_t)l * EMB + e) = v;
}

// ---------------------------------------------------------------------------
// Kernel B: xz = embeds @ Wx + b  for both directions, full fp32 via
// V_WMMA_F32_16X16X4_F32. Block = 256 thr = 8 waves, each wave owns a 16x16
// C tile; block tile = 64(M) x 32(N). K=512 in steps of 4.
// Fragment layouts per CDNA5 ISA 7.12.2 (32-bit A 16x4, 32-bit C/D 16x16).
// ---------------------------------------------------------------------------
__global__ void xz_gemm_wmma(const float* __restrict__ A,
                             const float* __restrict__ WxF, const float* __restrict__ bF,
                             const float* __restrict__ WxB, const float* __restrict__ bB,
                             float* __restrict__ xzF, float* __restrict__ xzB) {
  const int dir = blockIdx.z;
  const float* __restrict__ B    = dir ? WxB : WxF;
  const float* __restrict__ bias = dir ? bB  : bF;
  float* __restrict__ C          = dir ? xzB : xzF;

  const int tid  = threadIdx.x;
  const int w    = tid >> 5;
  const int lane = tid & 31;
  const int lr   = lane & 15;        // row/col within tile
  const int hi   = lane >> 4;        // lane half -> K offset 0 or 2
  const int m0   = blockIdx.x * 64 + (w & 3) * 16;
  const int n0   = blockIdx.y * 32 + (w >> 2) * 16;

  const int row = m0 + lr;           // A: M = lane%16
  const int col = n0 + lr;           // B/C: N = lane%16
  const v2f* __restrict__ A2 = reinterpret_cast<const v2f*>(A + (size_t)row * EMB);

  v8f acc = {0.f, 0.f, 0.f, 0.f, 0.f, 0.f, 0.f, 0.f};
  for (int k = 0; k < EMB; k += 4) {
    // A frag: v0 = K = k+2*hi, v1 = K+1  (contiguous -> one b64 load)
    v2f a = A2[(k >> 1) + hi];
    const int kk = k + 2 * hi;
    v2f b;
    b.x = B[(size_t)kk * GATES + col];
    b.y = B[(size_t)(kk + 1) * GATES + col];
    acc = __builtin_amdgcn_wmma_f32_16x16x4_f32(false, a, false, b,
                                                (short)0, acc, false, false);
  }
  const float bv = bias[col];
  #pragma unroll
  for (int r = 0; r < 8; ++r) {                        // C: VGPR r -> row r / r+8
    C[(size_t)(m0 + r + 8 * hi) * GATES + col] = acc[r] + bv;
  }
}

// ---------------------------------------------------------------------------
// Kernel C: persistent bidirectional LSTM scan.
// 64 blocks total: blocks [0,32) = forward, [32,64) = backward.
// Each block owns UPB=16 units => CPB=64 gate-columns of Wh held in LDS
// (transposed, padded). Per step: 64 dots of length 512 (4-way row split over
// 256 threads), local gate math, broadcast 16 h values via L2, monotonic
// atomic-counter barrier, re-read full h (double-buffered by step parity).
// ---------------------------------------------------------------------------
__global__ void bilstm_scan(const float* __restrict__ xzF, const float* __restrict__ xzB,
                            const float* __restrict__ WhF, const float* __restrict__ WhB,
                            const float* __restrict__ cF0, const float* __restrict__ hF0,
                            const float* __restrict__ cB0, const float* __restrict__ hB0,
                            float* __restrict__ hbuf,     // [2 parity][2 dir][512]
                            unsigned* __restrict__ cnt) { // [2]
  extern __shared__ float smem[];
  float* whs = smem;                          // CPB * WHS_STRIDE
  float* hsh = whs + CPB * WHS_STRIDE;        // 512
  float* zp  = hsh + UNITS;                   // 256 partials
  float* zf  = zp + 256;                      // 64 final z
  float* cst = zf + CPB;                      // 16 cell states

  const int tid = threadIdx.x;
  const int dir = blockIdx.x >> 5;
  const int bid = blockIdx.x & 31;

  const float* __restrict__ xz = dir ? xzB : xzF;
  const float* __restrict__ Wh = dir ? WhB : WhF;
  const float* __restrict__ c0 = dir ? cB0 : cF0;
  const float* __restrict__ h0 = dir ? hB0 : hF0;

  // Stage this block's Wh slice transposed into LDS: whs[c][r] = Wh[r][gcol(c)]
  for (int i = tid; i < CPB * UNITS; i += 256) {
    const int c = i >> 9;                     // column 0..63
    const int r = i & 511;                    // h index 0..511
    const int g = c >> 4, lu = c & 15;
    const int gcol = g * UNITS + bid * UPB + lu;
    whs[c * WHS_STRIDE + r] = Wh[(size_t)r * GATES + gcol];
  }
  hsh[tid]       = h0[tid];
  hsh[tid + 256] = h0[tid + 256];
  if (tid < UPB) cst[tid] = c0[bid * UPB + tid];
  __syncthreads();

  const int c      = tid & 63;
  const int q      = tid >> 6;                // 4-way row split
  const int rbase  = q * 128;
  const int g      = c >> 4, lu = c & 15;
  const int gcol   = g * UNITS + bid * UPB + lu;

  for (int t = 0; t < SEQ; ++t) {
    const int rrow = dir ? (SEQ - 1 - t) : t; // backward scans reversed rows

    // z partial: dot(h, Wh[:,c]) over 128 rows (hsh reads are wave-uniform)
    float acc = 0.f;
    const float* wcol = whs + c * WHS_STRIDE + rbase;
    const float* hh   = hsh + rbase;
    #pragma unroll 8
    for (int r = 0; r < 128; ++r) acc = fmaf(hh[r], wcol[r], acc);
    zp[tid] = acc;
    __syncthreads();

    if (tid < CPB)
      zf[tid] = zp[tid] + zp[tid + 64] + zp[tid + 128] + zp[tid + 192]
              + xz[(size_t)rrow * GATES + gcol];
    __syncthreads();

    const int pb = ((t + 1) & 1) * (2 * UNITS);   // parity double-buffer
    if (tid < UPB) {
      const float zi = zf[tid],      zfg = zf[16 + tid];
      const float zg = zf[32 + tid], zo  = zf[48 + tid];
      const float ig = 1.f / (1.f + expf(-zi));
      const float fg = 1.f / (1.f + expf(-zfg));
      const float gg = tanhf(zg);
      const float og = 1.f / (1.f + expf(-zo));
      const float cn = fg * cst[tid] + ig * gg;
      cst[tid] = cn;
      const float hn = og * tanhf(cn);
      __hip_atomic_store(&hbuf[pb + dir * UNITS + bid * UPB + tid], hn,
                         __ATOMIC_RELAXED, __HIP_MEMORY_SCOPE_AGENT);
    }
    __syncthreads();
    __builtin_amdgcn_fence(__ATOMIC_RELEASE, "agent");   // flush h slice to L2

    if (tid == 0) {
      __hip_atomic_fetch_add(&cnt[dir], 1u, __ATOMIC_RELEASE, __HIP_MEMORY_SCOPE_AGENT);
      const unsigned tgt = (unsigned)(DIRBLK * (t + 1));
      while (__hip_atomic_load(&cnt[dir], __ATOMIC_ACQUIRE, __HIP_MEMORY_SCOPE_AGENT) < tgt)
        __builtin_amdgcn_s_sleep(1);
    }
    __syncthreads();
    __builtin_amdgcn_fence(__ATOMIC_ACQUIRE, "agent");   // invalidate stale h lines

    hsh[tid]       = __hip_atomic_load(&hbuf[pb + dir * UNITS + tid],
                                       __ATOMIC_RELAXED, __HIP_MEMORY_SCOPE_AGENT);
    hsh[tid + 256] = __hip_atomic_load(&hbuf[pb + dir * UNITS + tid + 256],
                                       __ATOMIC_RELAXED, __HIP_MEMORY_SCOPE_AGENT);
    __syncthreads();
  }
  // Final h (t=2048, even parity) lives in hbuf[0..1023] for the merger.
}

// ---------------------------------------------------------------------------
// Kernel D: merged = tanh(concat(fwd_h, bwd_h) @ W + b)
// ---------------------------------------------------------------------------
__global__ void merger(const float* __restrict__ hbuf,   // parity-0: [fwd 512 | bwd 512]
                       const float* __restrict__ W, const float* __restrict__ b,
                       float* __restrict__ out) {
  __shared__ float hc[2 * UNITS];
  const int tid = threadIdx.x;
  for (int i = tid; i < 2 * UNITS; i += 256) hc[i] = hbuf[i];
  __syncthreads();
  for (int n = tid; n < UNITS; n += 256) {
    float acc = b[n];
    for (int j = 0; j < 2 * UNITS; ++j) acc = fmaf(hc[j], W[(size_t)j * UNITS + n], acc);
    out[n] = tanhf(acc);
  }
}

// ---------------------------------------------------------------------------
extern "C" void kernel_launch(void* const* d_in, const int* in_sizes, int n_in,
                              void* d_out, int out_size, void* d_ws, size_t ws_size,
                              hipStream_t stream) {
  (void)in_sizes; (void)n_in; (void)out_size; (void)ws_size;

  const int*   idx    = (const int*)  d_in[0];
  const float* vocab  = (const float*)d_in[1];
  const float* fwd_Wx = (const float*)d_in[2];
  const float* fwd_Wh = (const float*)d_in[3];
  const float* fwd_b  = (const float*)d_in[4];
  const float* bwd_Wx = (const float*)d_in[5];
  const float* bwd_Wh = (const float*)d_in[6];
  const float* bwd_b  = (const float*)d_in[7];
  const float* cF0    = (const float*)d_in[8];
  const float* hF0    = (const float*)d_in[9];
  const float* cB0    = (const float*)d_in[10];
  const float* hB0    = (const float*)d_in[11];
  const float* mW     = (const float*)d_in[12];
  const float* mb     = (const float*)d_in[13];
  float*       out    = (float*)d_out;

  // Workspace layout
  char*     ws     = (char*)d_ws;
  unsigned* cnt    = (unsigned*)ws;                 // 2 counters (reset every call)
  float*    base   = (float*)(ws + 256);
  float*    embeds = base;                          // 2048*512
  float*    xzF    = embeds + (size_t)SEQ * EMB;    // 2048*2048
  float*    xzB    = xzF + (size_t)SEQ * GATES;     // 2048*2048
  float*    hbuf   = xzB + (size_t)SEQ * GATES;     // 2*2*512

  hipMemsetAsync(cnt, 0, 2 * sizeof(unsigned), stream);

  gather_embeds<<<SEQ, 128, 0, stream>>>(idx, vocab, embeds);

  xz_gemm_wmma<<<dim3(SEQ / 64, GATES / 32, 2), 256, 0, stream>>>(
      embeds, fwd_Wx, fwd_b, bwd_Wx, bwd_b, xzF, xzB);

  const size_t lds = (size_t)(CPB * WHS_STRIDE + UNITS + 256 + CPB + UPB) * sizeof(float);
  hipFuncSetAttribute((const void*)bilstm_scan,
                      hipFuncAttributeMaxDynamicSharedMemorySize, (int)lds);
  bilstm_scan<<<2 * DIRBLK, 256, lds, stream>>>(
      xzF, xzB, fwd_Wh, bwd_Wh, cF0, hF0, cB0, hB0, hbuf, cnt);

  merger<<<1, 256, 0, stream>>>(hbuf, mW, mb, out);
}